// YoloXWrapper_TRT_60816736911852
// MI455X (gfx1250) — compile-verified
//
#include <hip/hip_runtime.h>
#include <stdint.h>

// ---------------- problem constants (from reference) ----------------
#define B_      64
#define N_      8400
#define C_      85
#define NCLS    80
#define TOPK_   200
#define INV639  (1.0f / 639.0f)   // H-1 == W-1 == 639

// ---------------- kernel-1 tiling ----------------
#define TILE_A      192                       // anchors per tile (mult of 4 -> 16B chunks)
#define K1_THREADS  256
#define NUM_TILES   ((N_ + TILE_A - 1) / TILE_A)   // 44 (last tile = 144 anchors)

#define K2_THREADS  256

// ---------------- CDNA5 async global->LDS path ----------------
#if defined(__AMDGCN__) && __has_builtin(__builtin_amdgcn_global_load_async_to_lds_b128)
#define USE_ASYNC 1
#else
#define USE_ASYNC 0
#endif

// Builtin signature (from hipcc diagnostic): pointee is GCC-vector int4.
typedef int v4i_ __attribute__((vector_size(4 * sizeof(int))));

__device__ __forceinline__ void copy16_g2lds(const float* __restrict__ gsrc,
                                             float* __restrict__ lds_dst) {
#if USE_ASYNC
  // (global src, lds dst, imm offset, imm cpol). Flat->AS3 via uintptr_t keeps
  // the low-32 LDS offset (aperture stripped by truncation).
  __builtin_amdgcn_global_load_async_to_lds_b128(
      reinterpret_cast<__attribute__((address_space(1))) v4i_*>(
          reinterpret_cast<uintptr_t>(gsrc)),
      reinterpret_cast<__attribute__((address_space(3))) v4i_*>(
          reinterpret_cast<uintptr_t>(lds_dst)),
      0, 0);
#else
  *reinterpret_cast<float4*>(lds_dst) = *reinterpret_cast<const float4*>(gsrc);
#endif
}

__device__ __forceinline__ void wait_async_zero() {
#if USE_ASYNC
#if __has_builtin(__builtin_amdgcn_s_wait_asynccnt)
  __builtin_amdgcn_s_wait_asynccnt(0);
#else
  asm volatile("s_wait_asynccnt 0" ::: "memory");
#endif
#endif
}

// =====================================================================
// Kernel 1: scores + argmax class.  Async-DMA a contiguous 192-anchor
// tile of det into LDS, then reduce 80 classes per anchor from LDS.
// =====================================================================
__global__ __launch_bounds__(K1_THREADS)
void yolox_score_kernel(const float* __restrict__ det,
                        float* __restrict__ scores,   // [B*N]
                        float* __restrict__ clsf) {   // [B*N] class index as float
  __shared__ __align__(16) float tile[TILE_A * C_];   // 65,280 B

  const int b  = blockIdx.y;
  const int t0 = blockIdx.x * TILE_A;
  int nA = N_ - t0;
  if (nA > TILE_A) nA = TILE_A;

  const float* __restrict__ src = det + ((size_t)b * N_ + (size_t)t0) * C_;
  const int nChunks = (nA * C_) >> 2;                 // 16-byte chunks (nA mult of 4)

  for (int j = threadIdx.x; j < nChunks; j += K1_THREADS)
    copy16_g2lds(src + j * 4, tile + j * 4);

  wait_async_zero();      // my wave's ASYNCcnt == 0  (data landed in LDS)
  __syncthreads();        // every wave's data landed

  if ((int)threadIdx.x < nA) {
    const float* row = tile + threadIdx.x * C_;       // stride 85 dwords: conflict-free
    const float obj  = row[4];
    float best = row[5];
    int   bi   = 0;
    for (int c = 1; c < NCLS; ++c) {
      const float v = row[5 + c];
      if (v > best) { best = v; bi = c; }             // strict '>' == first-max (jnp.argmax)
    }
    const size_t n = (size_t)b * N_ + (size_t)(t0 + (int)threadIdx.x);
    scores[n] = obj * best;
    clsf[n]   = (float)bi;
  }
}

// =====================================================================
// Kernel 2: per-batch top-200 (stable, descending), gather + emit.
// key64 = (float_bits(score)+1) << 32 | (8400 - n): scores >= 0 so the
// bit pattern is order-preserving; +1 reserves 0 as "removed"; low bits
// make ties pick the smallest anchor index (jax.lax.top_k order).
// =====================================================================
__device__ __forceinline__ unsigned long long shfl_xor_u64(unsigned long long v, int m) {
  unsigned int lo = (unsigned int)v;
  unsigned int hi = (unsigned int)(v >> 32);
  lo = __shfl_xor(lo, m, 32);
  hi = __shfl_xor(hi, m, 32);
  return ((unsigned long long)hi << 32) | lo;
}

__global__ __launch_bounds__(K2_THREADS)
void yolox_topk_kernel(const float* __restrict__ det,
                       const float* __restrict__ scores,
                       const float* __restrict__ clsf,
                       float* __restrict__ out,       // [B*TOPK, 7]
                       float* __restrict__ out_ids) { // [B*TOPK]
  __shared__ unsigned int       keyb[N_];             // 33,600 B
  __shared__ unsigned long long wmax[K2_THREADS / 32];
  __shared__ unsigned long long winner;
  __shared__ unsigned long long sel[TOPK_];

  const int b    = blockIdx.x;
  const int tid  = threadIdx.x;
  const int lane = tid & 31;
  const int wid  = tid >> 5;

  const float* __restrict__ sc = scores + (size_t)b * N_;

  if (tid < TOPK_) sel[tid] = 0ull;

  unsigned long long localKey = 0ull;
  for (int n = tid; n < N_; n += K2_THREADS) {
    const unsigned int k32 = __float_as_uint(sc[n]) + 1u;   // scores in [0,1): no overflow
    keyb[n] = k32;
    const unsigned long long k =
        ((unsigned long long)k32 << 32) | (unsigned int)(N_ - n);
    if (k > localKey) localKey = k;
  }
  __syncthreads();

  for (int i = 0; i < TOPK_; ++i) {
    // wave32 butterfly max
    unsigned long long k = localKey;
    for (int off = 16; off >= 1; off >>= 1) {
      const unsigned long long o = shfl_xor_u64(k, off);
      if (o > k) k = o;
    }
    if (lane == 0) wmax[wid] = k;
    __syncthreads();

    if (wid == 0) {
      unsigned long long m = (lane < (K2_THREADS / 32)) ? wmax[lane] : 0ull;
      for (int off = 4; off >= 1; off >>= 1) {
        const unsigned long long o = shfl_xor_u64(m, off);
        if (o > m) m = o;
      }
      if (lane == 0) winner = m;
    }
    __syncthreads();

    const unsigned long long w = winner;
    const int nwin = N_ - (int)(unsigned int)(w & 0xffffffffull);
    if ((nwin & (K2_THREADS - 1)) == tid) {            // owner of that anchor's slot
      sel[i]     = w;
      keyb[nwin] = 0u;                                 // remove
      unsigned long long nk = 0ull;                    // rescan my 33 slots
      for (int n = tid; n < N_; n += K2_THREADS) {
        const unsigned long long kk =
            ((unsigned long long)keyb[n] << 32) | (unsigned int)(N_ - n);
        if (keyb[n] != 0u && kk > nk) nk = kk;
      }
      localKey = nk;
    }
    __syncthreads();
  }

  // ---- emit (12800 x 7) + batch-id column ----
  for (int i = tid; i < TOPK_; i += K2_THREADS) {
    const unsigned long long w = sel[i];
    const int   n     = N_ - (int)(unsigned int)(w & 0xffffffffull);
    const float score = __uint_as_float((unsigned int)(w >> 32) - 1u);
    const float* __restrict__ drow = det + ((size_t)b * N_ + (size_t)n) * C_;
    const float cx = drow[0], cy = drow[1], ww = drow[2], hh = drow[3];
    const float cls = clsf[(size_t)b * N_ + (size_t)n];

    const int row = b * TOPK_ + i;
    float* __restrict__ o = out + (size_t)row * 7;
    o[0] = (float)b;
    o[1] = cls;
    o[2] = score;
    o[3] = (cx - ww * 0.5f) * INV639;
    o[4] = (cy - hh * 0.5f) * INV639;
    o[5] = (cx + ww * 0.5f) * INV639;
    o[6] = (cy + hh * 0.5f) * INV639;
    out_ids[row] = (float)b;
  }
}

// =====================================================================
extern "C" void kernel_launch(void* const* d_in, const int* in_sizes, int n_in,
                              void* d_out, int out_size, void* d_ws, size_t ws_size,
                              hipStream_t stream) {
  (void)in_sizes; (void)n_in; (void)out_size; (void)ws_size;

  // d_in[0] = x (64,3,640,640): only its shape (640) matters -> never read it.
  const float* det = (const float*)d_in[1];

  float* scores = (float*)d_ws;                       // B*N floats
  float* clsf   = scores + (size_t)B_ * N_;           // B*N floats  (4.3 MB total)

  float* out     = (float*)d_out;                     // 12800*7
  float* out_ids = out + (size_t)B_ * TOPK_ * 7;      // 12800

  dim3 g1(NUM_TILES, B_);
  hipLaunchKernelGGL(yolox_score_kernel, g1, dim3(K1_THREADS), 0, stream,
                     det, scores, clsf);
  hipLaunchKernelGGL(yolox_topk_kernel, dim3(B_), dim3(K2_THREADS), 0, stream,
                     det, scores, clsf, out, out_ids);
}